// CrossViewAttention_48584670052376
// MI455X (gfx1250) — compile-verified
//
#include <hip/hip_runtime.h>
#include <math.h>

// ---------------- constants ----------------
constexpr int BATCH = 4, NVIEW = 6, FEAT = 256, DIM = 128;
constexpr int HEADS = 4, DH = 32;
constexpr int FHh = 28, FWw = 60, HW = FHh * FWw;      // 1680
constexpr int BHh = 25, BWw = 25, QLEN = BHh * BWw;    // 625
constexpr int KLEN = NVIEW * HW;                       // 10080
constexpr float LN_EPS = 1e-5f;

typedef float v2f __attribute__((ext_vector_type(2)));
typedef float v8f __attribute__((ext_vector_type(8)));

static __device__ __forceinline__ v8f wmma4(v2f a, v2f b, v8f c) {
  // D = A(16x4,f32) * B(4x16,f32) + C(16x16,f32)
  return __builtin_amdgcn_wmma_f32_16x16x4_f32(false, a, false, b, (short)0, c,
                                               false, false);
}

static __device__ __forceinline__ v8f vzero8() {
  v8f z = {0.f, 0.f, 0.f, 0.f, 0.f, 0.f, 0.f, 0.f};
  return z;
}

static __device__ __forceinline__ v2f ld2(const float* p) {
  v2f r = {p[0], p[1]};
  return r;
}

// Block-wide (128 threads) LayerNorm stats via LDS. Returns (mean, rstd).
static __device__ __forceinline__ float2 ln_stats(float v, float* sh, int c) {
  sh[c] = v;
  __syncthreads();
  for (int s = 64; s > 0; s >>= 1) {
    if (c < s) sh[c] += sh[c + s];
    __syncthreads();
  }
  float mean = sh[0] * (1.0f / DIM);
  __syncthreads();
  float d = v - mean;
  sh[c] = d * d;
  __syncthreads();
  for (int s = 64; s > 0; s >>= 1) {
    if (c < s) sh[c] += sh[c + s];
    __syncthreads();
  }
  float var = sh[0] * (1.0f / DIM);
  __syncthreads();
  return make_float2(mean, rsqrtf(var + LN_EPS));
}

// ---------------- K1: img_embed (24,128,1680) ----------------
__global__ __launch_bounds__(128) void k_img_embed(
    const float* __restrict__ I_inv, const float* __restrict__ E_inv,
    const float* __restrict__ img_w, float* __restrict__ img_embed) {
  int blk = blockIdx.x;  // bn*HW + pix
  int bn = blk / HW, pix = blk % HW;
  int py = pix / FWw, px = pix % FWw;
  float gx = ((float)px / (float)(FWw - 1)) * 480.0f;
  float gy = ((float)py / (float)(FHh - 1)) * 224.0f;
  const float* I = I_inv + bn * 9;
  const float* E = E_inv + bn * 16;
  float c0 = I[0] * gx + I[1] * gy + I[2];
  float c1 = I[3] * gx + I[4] * gy + I[5];
  float c2 = I[6] * gx + I[7] * gy + I[8];
  float d0 = E[0] * c0 + E[1] * c1 + E[2] * c2 + E[3];
  float d1 = E[4] * c0 + E[5] * c1 + E[6] * c2 + E[7];
  float d2 = E[8] * c0 + E[9] * c1 + E[10] * c2 + E[11];
  float d3 = E[12] * c0 + E[13] * c1 + E[14] * c2 + E[15];
  int c = threadIdx.x;
  const float* w = img_w + c * 4;
  float v = w[0] * d0 + w[1] * d1 + w[2] * d2 + w[3] * d3;
  __shared__ float sh[DIM];
  sh[c] = v * v;
  __syncthreads();
  for (int s = 64; s > 0; s >>= 1) {
    if (c < s) sh[c] += sh[c + s];
    __syncthreads();
  }
  float norm = sqrtf(sh[0]) + 1e-7f;
  img_embed[((size_t)bn * DIM + c) * HW + pix] = v / norm;
}

// ---------------- K2: bev_embed (128,625) ----------------
__global__ __launch_bounds__(128) void k_bev_embed(
    const float* __restrict__ bev_grid, const float* __restrict__ bev_w,
    const float* __restrict__ bev_b, float* __restrict__ bev_embed) {
  int pix = blockIdx.x;
  int c = threadIdx.x;
  float g0 = bev_grid[pix];
  float g1 = bev_grid[QLEN + pix];
  float v = bev_w[c * 2 + 0] * g0 + bev_w[c * 2 + 1] * g1 + bev_b[c];
  __shared__ float sh[DIM];
  sh[c] = v * v;
  __syncthreads();
  for (int s = 64; s > 0; s >>= 1) {
    if (c < s) sh[c] += sh[c + s];
    __syncthreads();
  }
  float norm = sqrtf(sh[0]) + 1e-7f;
  bev_embed[c * QLEN + pix] = v / norm;
}

// ---- K3: batched conv1x1 with fused BN+ReLU: out(bn,128,1680) ----
// out = W(128x256) @ relu(feat*scale+b) [+ residual, same layout]
// Block = 8 waves as 4 M-groups x 2 N-pairs; wave tile = 32x32 (2x2 frags).
template <bool HASRES>
__global__ __launch_bounds__(256) void k_conv_gemm(
    const float* __restrict__ feat, const float* __restrict__ W,
    const float* __restrict__ bng, const float* __restrict__ bnb,
    const float* __restrict__ residual, float* __restrict__ out) {
  const float invs = 0.99999499993750052f;  // 1/sqrt(1+1e-5)
  int bn = blockIdx.y;
  int wave = threadIdx.x >> 5;
  int lane = threadIdx.x & 31;
  int hi = lane >> 4;
  int l16 = lane & 15;
  int mb = (wave >> 1) * 32;                    // 0,32,64,96
  int nb = blockIdx.x * 64 + (wave & 1) * 32;   // col base
  int n0 = nb + l16, n1 = nb + 16 + l16;
  int n0c = n0 < HW ? n0 : HW - 1;
  int n1c = n1 < HW ? n1 : HW - 1;
  const float* fbase = feat + (size_t)bn * FEAT * HW;
  v8f acc00 = vzero8(), acc01 = vzero8(), acc10 = vzero8(), acc11 = vzero8();
  for (int k0 = 0; k0 < FEAT; k0 += 4) {
    int ka = k0 + hi * 2;
    v2f a0 = ld2(W + (size_t)(mb + l16) * FEAT + ka);
    v2f a1 = ld2(W + (size_t)(mb + 16 + l16) * FEAT + ka);
    float s0 = bng[ka] * invs, s1 = bng[ka + 1] * invs;
    float c0 = bnb[ka], c1 = bnb[ka + 1];
    const float* fr0 = fbase + (size_t)ka * HW;
    const float* fr1 = fbase + (size_t)(ka + 1) * HW;
    v2f b0 = {fmaxf(fr0[n0c] * s0 + c0, 0.f), fmaxf(fr1[n0c] * s1 + c1, 0.f)};
    v2f b1 = {fmaxf(fr0[n1c] * s0 + c0, 0.f), fmaxf(fr1[n1c] * s1 + c1, 0.f)};
    acc00 = wmma4(a0, b0, acc00);
    acc01 = wmma4(a0, b1, acc01);
    acc10 = wmma4(a1, b0, acc10);
    acc11 = wmma4(a1, b1, acc11);
  }
#pragma unroll
  for (int r = 0; r < 8; r++) {
    int m0 = mb + r + hi * 8;
    int m1 = m0 + 16;
    if (n0 < HW) {
      size_t i00 = ((size_t)bn * DIM + m0) * HW + n0;
      size_t i10 = ((size_t)bn * DIM + m1) * HW + n0;
      float v00 = acc00[r], v10 = acc10[r];
      if (HASRES) {
        v00 += residual[i00];
        v10 += residual[i10];
      }
      out[i00] = v00;
      out[i10] = v10;
    }
    if (n1 < HW) {
      size_t i01 = ((size_t)bn * DIM + m0) * HW + n1;
      size_t i11 = ((size_t)bn * DIM + m1) * HW + n1;
      float v01 = acc01[r], v11 = acc11[r];
      if (HASRES) {
        v01 += residual[i01];
        v11 += residual[i11];
      }
      out[i01] = v01;
      out[i11] = v11;
    }
  }
}

// ---------------- K4a: qin = LN(bev+x), xt = x^T ----------------
__global__ __launch_bounds__(128) void k_build_qin(
    const float* __restrict__ x, const float* __restrict__ bev,
    const float* __restrict__ g, const float* __restrict__ bta,
    float* __restrict__ qin, float* __restrict__ xt) {
  int blk = blockIdx.x;  // b*QLEN + p
  int b = blk / QLEN, p = blk % QLEN;
  int c = threadIdx.x;
  float xv = x[((size_t)b * DIM + c) * QLEN + p];
  float v = bev[c * QLEN + p] + xv;
  __shared__ float sh[DIM];
  float2 st = ln_stats(v, sh, c);
  size_t row = (size_t)blk * DIM + c;
  xt[row] = xv;
  qin[row] = (v - st.x) * st.y * g[c] + bta[c];
}

// ------- K4b: LN + transpose (bn,128,1680) -> (b*10080, 128) -------
__global__ __launch_bounds__(128) void k_ln_kv(const float* __restrict__ src,
                                               const float* __restrict__ g,
                                               const float* __restrict__ bta,
                                               float* __restrict__ dst) {
  int r = blockIdx.x;  // b*KLEN + n*HW + p
  int b = r / KLEN;
  int rem = r % KLEN;
  int n = rem / HW, p = rem % HW;
  int c = threadIdx.x;
  float v = src[((size_t)(b * NVIEW + n) * DIM + c) * HW + p];
  __shared__ float sh[DIM];
  float2 st = ln_stats(v, sh, c);
  dst[(size_t)r * DIM + c] = (v - st.x) * st.y * g[c] + bta[c];
}

// -------- generic WMMA GEMM: C = act(A @ W^T + bias [+ res]) --------
// A: (M,K) row-major; W: (N,K) row-major.
// Block = 8 waves as 2 M-groups x 4 N-pairs; wave tile = 32x32 (2x2 frags).
// grid: x = ceil(M/64), y = N/128
template <bool HASRES, int ACT>
__global__ __launch_bounds__(256) void k_gemm(const float* __restrict__ A,
                                              const float* __restrict__ W,
                                              const float* __restrict__ bias,
                                              const float* __restrict__ residual,
                                              float* __restrict__ C, int M,
                                              int N, int K) {
  int wave = threadIdx.x >> 5;
  int lane = threadIdx.x & 31;
  int hi = lane >> 4;
  int l16 = lane & 15;
  int mb = blockIdx.x * 64 + (wave >> 2) * 32;
  int nb = blockIdx.y * 128 + (wave & 3) * 32;
  int r0 = mb + l16;
  int r1 = mb + 16 + l16;
  if (r0 > M - 1) r0 = M - 1;
  if (r1 > M - 1) r1 = M - 1;
  int n0 = nb + l16, n1 = nb + 16 + l16;
  v8f acc00 = vzero8(), acc01 = vzero8(), acc10 = vzero8(), acc11 = vzero8();
  for (int k0 = 0; k0 < K; k0 += 4) {
    int ka = k0 + hi * 2;
    v2f a0 = ld2(A + (size_t)r0 * K + ka);
    v2f a1 = ld2(A + (size_t)r1 * K + ka);
    v2f b0 = ld2(W + (size_t)n0 * K + ka);
    v2f b1 = ld2(W + (size_t)n1 * K + ka);
    acc00 = wmma4(a0, b0, acc00);
    acc01 = wmma4(a0, b1, acc01);
    acc10 = wmma4(a1, b0, acc10);
    acc11 = wmma4(a1, b1, acc11);
  }
  float bias0 = bias[n0], bias1 = bias[n1];
#pragma unroll
  for (int r = 0; r < 8; r++) {
    int m0 = mb + r + hi * 8;
    int m1 = m0 + 16;
#pragma unroll
    for (int half = 0; half < 2; half++) {
      int m = half ? m1 : m0;
      if (m >= M) continue;
      float v0 = (half ? acc10[r] : acc00[r]) + bias0;
      float v1 = (half ? acc11[r] : acc01[r]) + bias1;
      if (HASRES) {
        v0 += residual[(size_t)m * N + n0];
        v1 += residual[(size_t)m * N + n1];
      }
      if (ACT == 1) {
        v0 = 0.5f * v0 * (1.0f + erff(v0 * 0.70710678118654752f));
        v1 = 0.5f * v1 * (1.0f + erff(v1 * 0.70710678118654752f));
      }
      C[(size_t)m * N + n0] = v0;
      C[(size_t)m * N + n1] = v1;
    }
  }
}

// -------- flash attention: one wave per (b, head, 16-query tile) --------
// Q,K,V: (b, len, 128) with head slice [h*32, h*32+32); out: (b, QLEN, 128)
__global__ __launch_bounds__(32) void k_attention(const float* __restrict__ Q,
                                                  const float* __restrict__ Km,
                                                  const float* __restrict__ V,
                                                  float* __restrict__ out) {
  const int QT = (QLEN + 15) / 16;  // 40
  int idx = blockIdx.x;
  int qt = idx % QT;
  idx /= QT;
  int h = idx % HEADS;
  int b = idx / HEADS;
  int lane = threadIdx.x & 31;
  int hi = lane >> 4;
  int l16 = lane & 15;
  int qrow = qt * 16 + l16;
  if (qrow > QLEN - 1) qrow = QLEN - 1;
  const float* qp = Q + ((size_t)b * QLEN + qrow) * DIM + h * DH;
  float qb0[8], qb1[8];
#pragma unroll
  for (int c = 0; c < 8; c++) {
    int ka = 4 * c + hi * 2;
    qb0[c] = qp[ka];
    qb1[c] = qp[ka + 1];
  }
  const float* kbase = Km + (size_t)b * KLEN * DIM + h * DH;
  const float* vbase = V + (size_t)b * KLEN * DIM + h * DH;
  const float rscale = 0.17677669529663687f;  // 1/sqrt(32)
  float Mrun = -1e30f, Lrun = 0.f;
  v8f olo = vzero8(), ohi = vzero8();
  for (int kb = 0; kb < KLEN / 16; kb++) {
    int keyb = kb * 16;
    // T = K_blk(16x32) @ Q^T(32x16)  -> T[key][q], q = lane%16
    v8f t = vzero8();
#pragma unroll
    for (int c = 0; c < 8; c++) {
      int ka = 4 * c + hi * 2;
      v2f a = ld2(kbase + (size_t)(keyb + l16) * DIM + ka);
      v2f bq = {qb0[c], qb1[c]};
      t = wmma4(a, bq, t);
    }
    float P[8];
    float cm = -1e30f;
#pragma unroll
    for (int r = 0; r < 8; r++) {
      P[r] = t[r] * rscale;
      cm = fmaxf(cm, P[r]);
    }
    cm = fmaxf(cm, __shfl_xor(cm, 16, 32));
    float Mnew = fmaxf(Mrun, cm);
    float corr = __expf(Mrun - Mnew);
    float csum = 0.f;
#pragma unroll
    for (int r = 0; r < 8; r++) {
      P[r] = __expf(P[r] - Mnew);
      csum += P[r];
    }
    csum += __shfl_xor(csum, 16, 32);
    Lrun = Lrun * corr + csum;
    Mrun = Mnew;
#pragma unroll
    for (int r = 0; r < 8; r++) {
      olo[r] *= corr;
      ohi[r] *= corr;
    }
    // Build B fragments of P^T (16 keys x 16 queries) from C-layout of T.
    float sP[8];
#pragma unroll
    for (int r = 0; r < 8; r++) sP[r] = __shfl_xor(P[r], 16, 32);
    float bv0[4], bv1[4];
    bv0[0] = hi ? sP[2] : P[0];
    bv1[0] = hi ? sP[3] : P[1];
    bv0[1] = hi ? sP[6] : P[4];
    bv1[1] = hi ? sP[7] : P[5];
    bv0[2] = hi ? P[2] : sP[0];
    bv1[2] = hi ? P[3] : sP[1];
    bv0[3] = hi ? P[6] : sP[4];
    bv1[3] = hi ? P[7] : sP[5];
    // O^T(32x16) += V^T(32x16keys) @ P^T(16keys x 16q), two 16x16 tiles
#pragma unroll
    for (int c2 = 0; c2 < 4; c2++) {
      int key0 = keyb + 4 * c2 + hi * 2;
      const float* vp0 = vbase + (size_t)key0 * DIM;
      const float* vp1 = vbase + (size_t)(key0 + 1) * DIM;
      v2f bf = {bv0[c2], bv1[c2]};
      v2f alo = {vp0[l16], vp1[l16]};
      olo = wmma4(alo, bf, olo);
      v2f ahi = {vp0[16 + l16], vp1[16 + l16]};
      ohi = wmma4(ahi, bf, ohi);
    }
  }
  float invL = 1.0f / Lrun;
  int q = qt * 16 + l16;
  if (q < QLEN) {
    float* op = out + ((size_t)b * QLEN + q) * DIM + h * DH;
#pragma unroll
    for (int r = 0; r < 8; r++) {
      int d = r + hi * 8;
      op[d] = olo[r] * invL;
      op[d + 16] = ohi[r] * invL;
    }
  }
}

// ---------------- LN over rows (R,128) ----------------
__global__ __launch_bounds__(128) void k_ln_rows(const float* __restrict__ in,
                                                 const float* __restrict__ g,
                                                 const float* __restrict__ bta,
                                                 float* __restrict__ out) {
  int row = blockIdx.x;
  int c = threadIdx.x;
  float v = in[(size_t)row * DIM + c];
  __shared__ float sh[DIM];
  float2 st = ln_stats(v, sh, c);
  out[(size_t)row * DIM + c] = (v - st.x) * st.y * g[c] + bta[c];
}

// ------- final LN + transpose store to (b,128,25,25) -------
__global__ __launch_bounds__(128) void k_ln_final(const float* __restrict__ in,
                                                  const float* __restrict__ g,
                                                  const float* __restrict__ bta,
                                                  float* __restrict__ out) {
  int row = blockIdx.x;  // b*QLEN + p
  int b = row / QLEN, p = row % QLEN;
  int c = threadIdx.x;
  float v = in[(size_t)row * DIM + c];
  __shared__ float sh[DIM];
  float2 st = ln_stats(v, sh, c);
  out[((size_t)b * DIM + c) * QLEN + p] = (v - st.x) * st.y * g[c] + bta[c];
}

// ---------------- launch ----------------
extern "C" void kernel_launch(void* const* d_in, const int* in_sizes, int n_in,
                              void* d_out, int out_size, void* d_ws,
                              size_t ws_size, hipStream_t stream) {
  const float* x = (const float*)d_in[0];
  const float* feature = (const float*)d_in[1];
  const float* I_inv = (const float*)d_in[2];
  const float* E_inv = (const float*)d_in[3];
  const float* bev_grid = (const float*)d_in[4];
  const float* fl_bn_g = (const float*)d_in[5];
  const float* fl_bn_b = (const float*)d_in[6];
  const float* fl_conv_w = (const float*)d_in[7];
  const float* fp_bn_g = (const float*)d_in[8];
  const float* fp_bn_b = (const float*)d_in[9];
  const float* fp_conv_w = (const float*)d_in[10];
  const float* bev_w = (const float*)d_in[11];
  const float* bev_b = (const float*)d_in[12];
  const float* img_w = (const float*)d_in[13];
  const float* q_ln_g = (const float*)d_in[14];
  const float* q_ln_b = (const float*)d_in[15];
  const float* q_w = (const float*)d_in[16];
  const float* q_b = (const float*)d_in[17];
  const float* k_ln_g = (const float*)d_in[18];
  const float* k_ln_b = (const float*)d_in[19];
  const float* k_w = (const float*)d_in[20];
  const float* k_b = (const float*)d_in[21];
  const float* v_ln_g = (const float*)d_in[22];
  const float* v_ln_b = (const float*)d_in[23];
  const float* v_w = (const float*)d_in[24];
  const float* v_b = (const float*)d_in[25];
  const float* proj_w = (const float*)d_in[26];
  const float* proj_b = (const float*)d_in[27];
  const float* pre_g = (const float*)d_in[28];
  const float* pre_b = (const float*)d_in[29];
  const float* mlp1_w = (const float*)d_in[30];
  const float* mlp1_b = (const float*)d_in[31];
  const float* mlp2_w = (const float*)d_in[32];
  const float* mlp2_b = (const float*)d_in[33];
  const float* post_g = (const float*)d_in[34];
  const float* post_b = (const float*)d_in[35];

  // workspace layout (floats)
  const size_t BIG = (size_t)BATCH * NVIEW * DIM * HW;  // 5,160,960
  const size_t ROWS_Q = (size_t)BATCH * QLEN;           // 2500
  float* ws = (float*)d_ws;
  float* bufA = ws;          // img_embed -> kin -> vin
  float* bufB = bufA + BIG;  // key_flat -> kout
  float* bufC = bufB + BIG;  // val_flat -> vout
  float* bev = bufC + BIG;   // (128,625)
  float* qin = bev + (size_t)DIM * QLEN;
  float* xt = qin + ROWS_Q * DIM;
  float* qout = xt + ROWS_Q * DIM;
  float* aout = qout + ROWS_Q * DIM;
  float* zbuf = aout + ROWS_Q * DIM;
  float* z2 = zbuf + ROWS_Q * DIM;
  float* hdn = z2 + ROWS_Q * DIM;
  float* z3 = hdn + ROWS_Q * 2 * DIM;
  (void)in_sizes;
  (void)n_in;
  (void)out_size;
  (void)ws_size;

  const int MQ = BATCH * QLEN;  // 2500
  const int MK = BATCH * KLEN;  // 40320
  const int GX_Q = (MQ + 63) / 64;   // 40
  const int GX_K = MK / 64;          // 630
  const int GX_HW = (HW + 63) / 64;  // 27

  // 1. img_embed -> bufA
  k_img_embed<<<BATCH * NVIEW * HW, DIM, 0, stream>>>(I_inv, E_inv, img_w,
                                                      bufA);
  // 2. bev_embed
  k_bev_embed<<<QLEN, DIM, 0, stream>>>(bev_grid, bev_w, bev_b, bev);
  // 3. key_flat = img_embed + fp_conv(bn_relu(feat)) -> bufB
  k_conv_gemm<true><<<dim3(GX_HW, BATCH * NVIEW), 256, 0, stream>>>(
      feature, fp_conv_w, fp_bn_g, fp_bn_b, bufA, bufB);
  // 4. val_flat = fl_conv(bn_relu(feat)) -> bufC
  k_conv_gemm<false><<<dim3(GX_HW, BATCH * NVIEW), 256, 0, stream>>>(
      feature, fl_conv_w, fl_bn_g, fl_bn_b, nullptr, bufC);
  // 5. qin = LN(bev + x), xt
  k_build_qin<<<MQ, DIM, 0, stream>>>(x, bev, q_ln_g, q_ln_b, qin, xt);
  // 6. qout = qin @ q_w^T + q_b
  k_gemm<false, 0><<<dim3(GX_Q, 1), 256, 0, stream>>>(qin, q_w, q_b, nullptr,
                                                      qout, MQ, DIM, DIM);
  // 7. kin = LN(key_flat^T) -> bufA ; kout -> bufB
  k_ln_kv<<<MK, DIM, 0, stream>>>(bufB, k_ln_g, k_ln_b, bufA);
  k_gemm<false, 0><<<dim3(GX_K, 1), 256, 0, stream>>>(bufA, k_w, k_b, nullptr,
                                                      bufB, MK, DIM, DIM);
  // 8. vin = LN(val_flat^T) -> bufA ; vout -> bufC
  k_ln_kv<<<MK, DIM, 0, stream>>>(bufC, v_ln_g, v_ln_b, bufA);
  k_gemm<false, 0><<<dim3(GX_K, 1), 256, 0, stream>>>(bufA, v_w, v_b, nullptr,
                                                      bufC, MK, DIM, DIM);
  // 9. attention
  k_attention<<<BATCH * HEADS * ((QLEN + 15) / 16), 32, 0, stream>>>(
      qout, bufB, bufC, aout);
  // 10. z = a @ proj_w^T + proj_b + x^T
  k_gemm<true, 0><<<dim3(GX_Q, 1), 256, 0, stream>>>(aout, proj_w, proj_b, xt,
                                                     zbuf, MQ, DIM, DIM);
  // 11. z2 = LN(z)
  k_ln_rows<<<MQ, DIM, 0, stream>>>(zbuf, pre_g, pre_b, z2);
  // 12. hdn = gelu(z2 @ mlp1_w^T + mlp1_b)  (N=256)
  k_gemm<false, 1><<<dim3(GX_Q, 2), 256, 0, stream>>>(z2, mlp1_w, mlp1_b,
                                                      nullptr, hdn, MQ,
                                                      2 * DIM, DIM);
  // 13. z3 = z2 + hdn @ mlp2_w^T + mlp2_b  (K=256)
  k_gemm<true, 0><<<dim3(GX_Q, 1), 256, 0, stream>>>(hdn, mlp2_w, mlp2_b, z2,
                                                     z3, MQ, DIM, 2 * DIM);
  // 14. out = LN(z3)^T -> (b,128,25,25)
  k_ln_final<<<MQ, DIM, 0, stream>>>(z3, post_g, post_b, (float*)d_out);
}